// MyModel_61933428410370
// MI455X (gfx1250) — compile-verified
//
#include <hip/hip_runtime.h>
#include <hip/hip_bf16.h>
#include <limits.h>

// Native clang vector so __builtin_nontemporal_load accepts it (HIP's float4
// is a class type and is rejected by the builtin).
typedef float v4f __attribute__((ext_vector_type(4)));

// Per-block partial result: max value, first index at max, last index at max.
struct ArgPack {
    float v;
    int   lo;
    int   hi;
    int   pad;   // 16-byte aligned for clean b128 traffic on the partials
};

// Tie-aware combine: larger value wins; equal value merges index ranges.
__device__ __forceinline__ void combine(float& v, int& lo, int& hi,
                                        float v2, int lo2, int hi2) {
    if (v2 > v) {
        v = v2; lo = lo2; hi = hi2;
    } else if (v2 == v) {
        lo = (lo2 < lo) ? lo2 : lo;
        hi = (hi2 > hi) ? hi2 : hi;
    }
}

// wave32 + cross-wave (LDS) reduction of (v, lo, hi) within a 256-thread block.
// Returns the block result in thread 0's (v, lo, hi).
__device__ __forceinline__ void block_reduce(float& v, int& lo, int& hi) {
    // wave32 shuffle reduction (CDNA5 is wave32-only; width must be 32).
    #pragma unroll
    for (int off = 16; off > 0; off >>= 1) {
        float v2  = __shfl_down(v,  off, 32);
        int   lo2 = __shfl_down(lo, off, 32);
        int   hi2 = __shfl_down(hi, off, 32);
        combine(v, lo, hi, v2, lo2, hi2);
    }

    __shared__ float sv[8];
    __shared__ int   slo[8];
    __shared__ int   shi[8];

    const int lane = threadIdx.x & 31;
    const int wid  = threadIdx.x >> 5;   // 256 threads -> 8 waves

    if (lane == 0) { sv[wid] = v; slo[wid] = lo; shi[wid] = hi; }
    __syncthreads();

    if (wid == 0) {
        const int nw = blockDim.x >> 5;  // number of waves (8)
        v  = (lane < nw) ? sv[lane]  : -__builtin_huge_valf();
        lo = (lane < nw) ? slo[lane] : INT_MAX;
        hi = (lane < nw) ? shi[lane] : -1;
        #pragma unroll
        for (int off = 4; off > 0; off >>= 1) {
            float v2  = __shfl_down(v,  off, 32);
            int   lo2 = __shfl_down(lo, off, 32);
            int   hi2 = __shfl_down(hi, off, 32);
            combine(v, lo, hi, v2, lo2, hi2);
        }
    }
}

// Pass 1: grid-stride streaming argmax with tie range tracking.
// NT b128 loads: 256 MB streamed once, larger than the 192 MB L2 -> don't cache.
__global__ void argmax_partial_kernel(const float* __restrict__ x, int n,
                                      ArgPack* __restrict__ part) {
    const int tid   = threadIdx.x;
    const int gid   = blockIdx.x * blockDim.x + tid;
    const int gsize = gridDim.x * blockDim.x;
    const int n4    = n >> 2;

    const v4f* __restrict__ x4 = (const v4f*)x;

    float v  = -__builtin_huge_valf();
    int   lo = INT_MAX;
    int   hi = -1;

    for (int i = gid; i < n4; i += gsize) {
        // Prefetch the next grid-stride iteration's line (speculative;
        // out-of-range translation failures are silently dropped).
        __builtin_prefetch(&x4[i + gsize], 0, 1);
        const v4f d = __builtin_nontemporal_load(&x4[i]);
        const int base = i << 2;
        combine(v, lo, hi, d.x, base,     base);
        combine(v, lo, hi, d.y, base + 1, base + 1);
        combine(v, lo, hi, d.z, base + 2, base + 2);
        combine(v, lo, hi, d.w, base + 3, base + 3);
    }

    // Tail (n % 4 elements; n = 2^26 here so this is a no-op, kept for safety).
    const int t = (n4 << 2) + gid;
    if (t < n) {
        const float e = x[t];
        combine(v, lo, hi, e, t, t);
    }

    block_reduce(v, lo, hi);

    if (threadIdx.x == 0) {
        ArgPack p; p.v = v; p.lo = lo; p.hi = hi; p.pad = 0;
        part[blockIdx.x] = p;
    }
}

// Pass 2: single block folds the per-block partials; emits 1.0 iff the max
// value's first and last indices differ (i.e., the max is not unique).
__global__ void argmax_final_kernel(const ArgPack* __restrict__ part, int nparts,
                                    float* __restrict__ out) {
    float v  = -__builtin_huge_valf();
    int   lo = INT_MAX;
    int   hi = -1;

    for (int i = threadIdx.x; i < nparts; i += blockDim.x) {
        const ArgPack p = part[i];
        combine(v, lo, hi, p.v, p.lo, p.hi);
    }

    block_reduce(v, lo, hi);

    if (threadIdx.x == 0) {
        out[0] = (lo != hi) ? 1.0f : 0.0f;
    }
}

extern "C" void kernel_launch(void* const* d_in, const int* in_sizes, int n_in,
                              void* d_out, int out_size, void* d_ws, size_t ws_size,
                              hipStream_t stream) {
    const float* x    = (const float*)d_in[0];
    const int    n    = in_sizes[0];          // 64 * 1048576 = 2^26
    float*       out  = (float*)d_out;
    ArgPack*     part = (ArgPack*)d_ws;

    // 2048 blocks x 256 threads x 4 elem/b128-load -> 32 streaming iterations
    // per thread; enough in-flight b128 loads to saturate 23.3 TB/s HBM.
    int nblocks = 2048;
    const int maxblocks = (int)(ws_size / sizeof(ArgPack));
    if (nblocks > maxblocks) nblocks = maxblocks;
    if (nblocks < 1) nblocks = 1;

    argmax_partial_kernel<<<nblocks, 256, 0, stream>>>(x, n, part);
    argmax_final_kernel<<<1, 256, 0, stream>>>(part, nblocks, out);
}